// Model_20547123544421
// MI455X (gfx1250) — compile-verified
//
#include <hip/hip_runtime.h>

// ---------------------------------------------------------------------------
// Mamba-2 SSD chunked scan for gfx1250 (MI455X), fp32 WMMA path.
// Shapes (hardcoded from reference): b=4, s=8192, h=32, p=64, n=16, l=64, c=128
// Memory-bound (~1.2 GB min traffic @ 23.3 TB/s): all tile staging uses b128
// global loads + b128 LDS stores; LDS rows padded by +4 floats (16B-aligned,
// bank-conflict-free for both row and column fragment reads).
// ---------------------------------------------------------------------------

typedef float v2f __attribute__((ext_vector_type(2)));
typedef float v8f __attribute__((ext_vector_type(8)));

#define WMMA_F32_4(a, b, c) \
  __builtin_amdgcn_wmma_f32_16x16x4_f32(false, (a), false, (b), (short)0, (c), false, false)

#define NB 4
#define NS 8192
#define NH 32
#define NP 64
#define NN 16
#define LCH 64
#define NC 128

#define XPAD 4   // row stride 68 floats = 272B (16B aligned), banks (4m+k)%64 distinct
#define NPAD 4   // row stride 20 floats =  80B (16B aligned), banks (20m+k)%64 distinct

// Inclusive Hillis-Steele scan of 64 values living in shared memory.
// Caller must have written cs[0..63] and issued __syncthreads() already.
__device__ __forceinline__ void scan64(volatile float* cs, int tid) {
#pragma unroll
  for (int off = 1; off < 64; off <<= 1) {
    float v = 0.0f;
    if (tid < 64 && tid >= off) v = cs[tid - off];
    __syncthreads();
    if (tid < 64) cs[tid] += v;
    __syncthreads();
  }
}

// ---------------------------------------------------------------------------
// Kernel 1: per-chunk states  S[b,c,h,p,n] = sum_l exp(cs[63]-cs[l]) X[l,p] B[l,n]
// and A_last[b,h,c] = cs[63].
// ---------------------------------------------------------------------------
__global__ __launch_bounds__(128) void k1_states(const float* __restrict__ X,
                                                 const float* __restrict__ A,
                                                 const float* __restrict__ Bm,
                                                 float* __restrict__ S,
                                                 float* __restrict__ Alast) {
  __shared__ __align__(16) float Xs[LCH][NP + XPAD];
  __shared__ __align__(16) float Bs[LCH][NN + NPAD];
  __shared__ float cs[LCH];
  __shared__ float ds[LCH];

  const int tid = threadIdx.x;
  const int hG = blockIdx.x, cG = blockIdx.y, bG = blockIdx.z;

  // X tile: 64x64 f32 = 1024 float4, b128 coalesced (lane-consecutive 16B).
  const size_t xbase = ((size_t)((bG * NS + cG * LCH) * NH) + hG) * NP;
  const float4* X4 = (const float4*)(X + xbase);
  for (int vi = tid; vi < LCH * (NP / 4); vi += 128) {
    const int l = vi >> 4, q = vi & 15;
    *(float4*)&Xs[l][q * 4] = X4[(size_t)l * (NH * NP / 4) + q];
  }
  // B tile: 64x16 f32 = 256 float4.
  const size_t bbase = ((size_t)((bG * NS + cG * LCH) * NH) + hG) * NN;
  const float4* B4 = (const float4*)(Bm + bbase);
  for (int vi = tid; vi < LCH * (NN / 4); vi += 128) {
    const int l = vi >> 2, q = vi & 3;
    *(float4*)&Bs[l][q * 4] = B4[(size_t)l * (NH * NN / 4) + q];
  }
  const size_t abase = (size_t)(bG * NS + cG * LCH) * NH + hG;
  if (tid < 64) cs[tid] = A[abase + (size_t)tid * NH];
  __syncthreads();

  scan64(cs, tid);
  if (tid < 64) ds[tid] = __expf(cs[63] - cs[tid]);
  __syncthreads();

  const int w = tid >> 5;          // wave -> p tile
  const int lane = tid & 31;
  const int m = lane & 15;
  const int kh = lane >> 4;

  v8f acc = {};
#pragma unroll
  for (int kb = 0; kb < LCH; kb += 4) {
    const int k0 = kb + 2 * kh;
    v2f av, bv;
    av.x = ds[k0] * Xs[k0][w * 16 + m];           // A[p][l] = decay*X (transposed read)
    av.y = ds[k0 + 1] * Xs[k0 + 1][w * 16 + m];
    bv.x = Bs[k0][m];                             // B[l][n]
    bv.y = Bs[k0 + 1][m];
    acc = WMMA_F32_4(av, bv, acc);
  }

  const size_t sbase = ((size_t)((bG * NC + cG) * NH) + hG) * (NP * NN);
#pragma unroll
  for (int v = 0; v < 8; ++v) {
    const int p = w * 16 + v + 8 * kh;
    S[sbase + p * NN + m] = acc[v];
  }
  if (tid == 0) Alast[((size_t)bG * NH + hG) * NC + cG] = cs[63];
}

// ---------------------------------------------------------------------------
// Kernel 2: sequential inter-chunk scan (in place on S).
// out[c] = prev;  prev = prev*exp(A_last[c]) + S_in[c]
// 131072 independent (b,h,p,n) scan lanes: grid (8, 32, 4) x 128 threads,
// each block touching a contiguous 512B slice per step.
// ---------------------------------------------------------------------------
__global__ __launch_bounds__(128) void k2_scan(float* __restrict__ S,
                                               const float* __restrict__ Alast) {
  const int slice = blockIdx.x;    // 0..7, 128 elements each of the 1024 (p,n)
  const int hG = blockIdx.y, bG = blockIdx.z;
  const int e = slice * 128 + threadIdx.x;
  const float* al = Alast + ((size_t)bG * NH + hG) * NC;

  float prev = 0.0f;
  for (int c = 0; c < NC; ++c) {
    const size_t base = ((size_t)((bG * NC + c) * NH) + hG) * (NP * NN) + e;
    const float d = __expf(al[c]);
    const float val = S[base];
    S[base] = prev;
    prev = prev * d + val;
  }
}

// ---------------------------------------------------------------------------
// Kernel 3:  Y = (C B^T ∘ L) X  +  (C ∘ exp(cs)) S^T
// ---------------------------------------------------------------------------
__global__ __launch_bounds__(128) void k3_y(const float* __restrict__ X,
                                            const float* __restrict__ A,
                                            const float* __restrict__ Bm,
                                            const float* __restrict__ Cm,
                                            const float* __restrict__ S,
                                            float* __restrict__ Y) {
  __shared__ __align__(16) float Xs[LCH][NP + XPAD];
  __shared__ __align__(16) float Ws[LCH][LCH + XPAD];
  __shared__ __align__(16) float Bs[LCH][NN + NPAD];
  __shared__ __align__(16) float Csh[LCH][NN + NPAD];
  __shared__ __align__(16) float Ss[NP][NN + NPAD];
  __shared__ float cs[LCH];
  __shared__ float ecs[LCH];

  const int tid = threadIdx.x;
  const int hG = blockIdx.x, cG = blockIdx.y, bG = blockIdx.z;

  const size_t xbase = ((size_t)((bG * NS + cG * LCH) * NH) + hG) * NP;
  const float4* X4 = (const float4*)(X + xbase);
  for (int vi = tid; vi < LCH * (NP / 4); vi += 128) {
    const int l = vi >> 4, q = vi & 15;
    *(float4*)&Xs[l][q * 4] = X4[(size_t)l * (NH * NP / 4) + q];
  }
  const size_t bcbase = ((size_t)((bG * NS + cG * LCH) * NH) + hG) * NN;
  const float4* B4 = (const float4*)(Bm + bcbase);
  const float4* C4 = (const float4*)(Cm + bcbase);
  for (int vi = tid; vi < LCH * (NN / 4); vi += 128) {
    const int l = vi >> 2, q = vi & 3;
    *(float4*)&Bs[l][q * 4] = B4[(size_t)l * (NH * NN / 4) + q];
    *(float4*)&Csh[l][q * 4] = C4[(size_t)l * (NH * NN / 4) + q];
  }
  // States tile: 1024 contiguous floats = 256 float4.
  const size_t sbase = ((size_t)((bG * NC + cG) * NH) + hG) * (NP * NN);
  const float4* S4 = (const float4*)(S + sbase);
  for (int vi = tid; vi < NP * (NN / 4); vi += 128) {
    const int p = vi >> 2, q = vi & 3;
    *(float4*)&Ss[p][q * 4] = S4[vi];
  }
  const size_t abase = (size_t)(bG * NS + cG * LCH) * NH + hG;
  if (tid < 64) cs[tid] = A[abase + (size_t)tid * NH];
  __syncthreads();

  scan64(cs, tid);
  if (tid < 64) ecs[tid] = __expf(cs[tid]);
  __syncthreads();

  const int w = tid >> 5;          // wave -> row strip ti=w for phase 1
  const int lane = tid & 31;
  const int m = lane & 15;
  const int kh = lane >> 4;

  // ---- Phase 1: W = (C B^T) ∘ L, lower-triangular tiles only ----
  for (int tj = 0; tj <= w; ++tj) {
    v8f acc = {};
#pragma unroll
    for (int kb = 0; kb < NN; kb += 4) {
      const int k0 = kb + 2 * kh;
      v2f av, bv;
      av.x = Csh[w * 16 + m][k0];            // A[i][n]
      av.y = Csh[w * 16 + m][k0 + 1];
      bv.x = Bs[tj * 16 + m][k0];            // B^T[n][j] = B[j][n]
      bv.y = Bs[tj * 16 + m][k0 + 1];
      acc = WMMA_F32_4(av, bv, acc);
    }
#pragma unroll
    for (int v = 0; v < 8; ++v) {
      const int i = w * 16 + v + 8 * kh;
      const int j = tj * 16 + m;
      Ws[i][j] = (j <= i) ? acc[v] * __expf(cs[i] - cs[j]) : 0.0f;
    }
  }
  __syncthreads();

  // ---- Phase 2: Y tile (ti, tp) = C2 @ S^T + W @ X ----
  const size_t ybase = xbase;
#pragma unroll
  for (int t = 0; t < 4; ++t) {
    const int ti = t;
    const int tp = (w + t) & 3;              // balance triangular cost over waves
    v8f acc = {};
#pragma unroll
    for (int kb = 0; kb < NN; kb += 4) {     // Y_off: contract over n
      const int k0 = kb + 2 * kh;
      const float e = ecs[ti * 16 + m];
      v2f av, bv;
      av.x = Csh[ti * 16 + m][k0] * e;       // C2[i][n]
      av.y = Csh[ti * 16 + m][k0 + 1] * e;
      bv.x = Ss[tp * 16 + m][k0];            // S^T[n][p] = S[p][n]
      bv.y = Ss[tp * 16 + m][k0 + 1];
      acc = WMMA_F32_4(av, bv, acc);
    }
    for (int kb = 0; kb < (ti + 1) * 16; kb += 4) {  // W @ X, skip zero K-range
      const int k0 = kb + 2 * kh;
      v2f av, bv;
      av.x = Ws[ti * 16 + m][k0];
      av.y = Ws[ti * 16 + m][k0 + 1];
      bv.x = Xs[k0][tp * 16 + m];
      bv.y = Xs[k0 + 1][tp * 16 + m];
      acc = WMMA_F32_4(av, bv, acc);
    }
#pragma unroll
    for (int v = 0; v < 8; ++v) {
      const int i = ti * 16 + v + 8 * kh;
      const int p = tp * 16 + m;
      Y[ybase + (size_t)i * (NH * NP) + p] = acc[v];
    }
  }
}

// ---------------------------------------------------------------------------
extern "C" void kernel_launch(void* const* d_in, const int* in_sizes, int n_in,
                              void* d_out, int out_size, void* d_ws, size_t ws_size,
                              hipStream_t stream) {
  const float* X = (const float*)d_in[0];
  const float* A = (const float*)d_in[1];
  const float* Bm = (const float*)d_in[2];
  const float* Cm = (const float*)d_in[3];
  float* Y = (float*)d_out;

  // Workspace layout: [states: b*c*h*p*n f32 = 64 MiB][A_last: b*h*c f32 = 64 KiB]
  float* S = (float*)d_ws;
  float* Alast = S + (size_t)NB * NC * NH * NP * NN;

  dim3 gridBCH(NH, NC, NB);
  k1_states<<<gridBCH, 128, 0, stream>>>(X, A, Bm, S, Alast);
  k2_scan<<<dim3(8, NH, NB), 128, 0, stream>>>(S, Alast);
  k3_y<<<gridBCH, 128, 0, stream>>>(X, A, Bm, Cm, S, Y);
}